// Loss1_69355131896601
// MI455X (gfx1250) — compile-verified
//
#include <hip/hip_runtime.h>

// ---------------------------------------------------------------------------
// Masked-mean reduction:  out = sum(x-y | x>y) / count(x>y)   (f32, N = 33.5M)
//
// Memory-bound: 268 MB read @ 23.3 TB/s -> ~11.5 us floor. Strategy:
//   Pass 1: 1024 blocks x 256 threads, grid-stride float4 (b128) non-temporal
//           loads, per-thread {float sum, uint count}, wave32 shfl_xor
//           reduction, LDS across 8 waves, deterministic block partials to ws.
//   Pass 2: 1 wave (32 threads) reduces the 1024 partials. Count reduced
//           exactly via integer shfl_xor; float sum reduced with a chain of
//           V_WMMA_F32_16X16X4_F32 (A = per-lane partials per the documented
//           16x4 f32 A-layout, B = all-ones which is layout-independent):
//           D[m,n] = sum_k A[m,k], then 4 accumulating WMMAs collapse the 16
//           row-sums -> total in every element of D. Deterministic, exact
//           per ISA layout tables, and exercises the gfx1250 WMMA path.
// ---------------------------------------------------------------------------

typedef float f4  __attribute__((ext_vector_type(4)));
typedef float v2f __attribute__((ext_vector_type(2)));
typedef float v8f __attribute__((ext_vector_type(8)));

#define NBLOCKS  1024
#define NTHREADS 256

__global__ __launch_bounds__(NTHREADS)
void loss1_partial(const float* __restrict__ x, const float* __restrict__ y,
                   float* __restrict__ blkSum, unsigned* __restrict__ blkCnt, int n)
{
    const int tid    = blockIdx.x * blockDim.x + threadIdx.x;
    const int stride = gridDim.x * blockDim.x;
    const int n4     = n >> 2;

    const f4* __restrict__ x4 = (const f4*)x;
    const f4* __restrict__ y4 = (const f4*)y;

    float    s = 0.0f;
    unsigned c = 0u;

    // Streaming main loop: 128-bit non-temporal loads (268MB > 192MB L2).
    for (int i = tid; i < n4; i += stride) {
        f4 a = __builtin_nontemporal_load(&x4[i]);
        f4 b = __builtin_nontemporal_load(&y4[i]);
#pragma unroll
        for (int k = 0; k < 4; ++k) {
            float d = a[k] - b[k];
            if (d > 0.0f) { s += d; c += 1u; }
        }
    }
    // Scalar tail (empty for N % 4 == 0, kept for generality).
    for (int i = (n4 << 2) + tid; i < n; i += stride) {
        float d = x[i] - y[i];
        if (d > 0.0f) { s += d; c += 1u; }
    }

    // Wave32 butterfly reduction (deterministic).
#pragma unroll
    for (int off = 16; off > 0; off >>= 1) {
        s += __shfl_xor(s, off, 32);
        c += __shfl_xor(c, off, 32);
    }

    __shared__ float    ls[NTHREADS / 32];
    __shared__ unsigned lc[NTHREADS / 32];
    const int lane = threadIdx.x & 31;
    const int wave = threadIdx.x >> 5;
    if (lane == 0) { ls[wave] = s; lc[wave] = c; }
    __syncthreads();

    if (wave == 0) {
        float    ws = (lane < (NTHREADS / 32)) ? ls[lane] : 0.0f;
        unsigned wc = (lane < (NTHREADS / 32)) ? lc[lane] : 0u;
#pragma unroll
        for (int off = 4; off > 0; off >>= 1) {
            ws += __shfl_xor(ws, off, 32);
            wc += __shfl_xor(wc, off, 32);
        }
        if (lane == 0) { blkSum[blockIdx.x] = ws; blkCnt[blockIdx.x] = wc; }
    }
}

// Single-wave final reduction. EXEC is all-ones for the WMMA section
// (uniform control flow; divergence only after the matrix ops).
__global__ __launch_bounds__(32)
void loss1_final(const float* __restrict__ blkSum, const unsigned* __restrict__ blkCnt,
                 float* __restrict__ out)
{
    const int lane = threadIdx.x;  // 0..31, one full wave32

    float    s = 0.0f;
    unsigned c = 0u;
    // Uniform trip count (NBLOCKS/32 per lane): no divergence.
    for (int i = lane; i < NBLOCKS; i += 32) {
        s += blkSum[i];
        c += blkCnt[i];
    }

    // Exact integer count reduction.
#pragma unroll
    for (int off = 16; off > 0; off >>= 1) c += __shfl_xor(c, off, 32);

    // ---- WMMA-based float sum across the wave ----
    // A-matrix 16x4 f32 layout (ISA 7.12.2): lane L (<16) holds A[L][0],A[L][1]
    // in v0,v1; lane L+16 holds A[L][2],A[L][3]. We place each lane's partial
    // in v0, zero in v1, so row m of A holds {s_m, 0, s_{m+16}, 0}.
    v2f A; A.x = s;    A.y = 0.0f;
    v2f B; B.x = 1.0f; B.y = 1.0f;   // all-ones B: layout-independent
    v8f Z = {};

    // D1[m][n] = s_m + s_{m+16}  (replicated across all 16 columns).
    v8f D1 = __builtin_amdgcn_wmma_f32_16x16x4_f32(
        false, A, false, B, (short)0, Z, false, false);

    // D1 VGPR r holds rowsum_r (lanes 0-15) / rowsum_{r+8} (lanes 16-31).
    // Feeding VGPR pairs (2p, 2p+1) back as A sums rows {2p,2p+1,2p+8,2p+9};
    // accumulating over p=0..3 sums all 16 row-sums -> grand total everywhere.
    v8f acc = {};
#pragma unroll
    for (int p = 0; p < 4; ++p) {
        v2f Ap; Ap.x = D1[2 * p]; Ap.y = D1[2 * p + 1];
        acc = __builtin_amdgcn_wmma_f32_16x16x4_f32(
            false, Ap, false, B, (short)0, acc, false, false);
    }
    float total = acc[0];  // identical in every lane/element

    if (lane == 0) {
        float cf = (float)c;
        out[0] = (c > 0u) ? (total / fmaxf(cf, 1.0f)) : 0.0f;
    }
}

extern "C" void kernel_launch(void* const* d_in, const int* in_sizes, int n_in,
                              void* d_out, int out_size, void* d_ws, size_t ws_size,
                              hipStream_t stream)
{
    const float* x = (const float*)d_in[0];
    const float* y = (const float*)d_in[1];
    const int    n = in_sizes[0];

    float*    blkSum = (float*)d_ws;
    unsigned* blkCnt = (unsigned*)((char*)d_ws + NBLOCKS * sizeof(float));

    loss1_partial<<<NBLOCKS, NTHREADS, 0, stream>>>(x, y, blkSum, blkCnt, n);
    loss1_final<<<1, 32, 0, stream>>>(blkSum, blkCnt, (float*)d_out);
}